// NonMaxSuppression_22316650070073
// MI455X (gfx1250) — compile-verified
//
#include <hip/hip_runtime.h>

#define NMS_B 8
#define NMS_C 32
#define NMS_N 1000
#define NPAD  1008          // 63 tiles of 16
#define ORD_STRIDE 1024

typedef __attribute__((ext_vector_type(2))) float v2f;
typedef __attribute__((ext_vector_type(8))) float v8f;

// ---------------------------------------------------------------------------
// Kernel 1: stable descending argsort per (b,c) via O(N^2) rank counting.
// rank(e) = #{j : s[j] > s[e]} + #{j < e : s[j] == s[e]}  == jnp.argsort(-s)
// ---------------------------------------------------------------------------
__global__ void nms_sort_kernel(const float* __restrict__ scores,
                                int* __restrict__ order)
{
    __shared__ float s[NMS_N];
    const int bc = blockIdx.x;
    const float* sc = scores + (size_t)bc * NMS_N;
    for (int i = threadIdx.x; i < NMS_N; i += blockDim.x) s[i] = sc[i];
    __syncthreads();
    for (int e = threadIdx.x; e < NMS_N; e += blockDim.x) {
        const float se = s[e];
        int rank = 0;
        for (int j = 0; j < NMS_N; ++j) {
            const float sj = s[j];
            rank += (sj > se) || ((sj == se) && (j < e));
        }
        order[bc * ORD_STRIDE + rank] = e;
    }
}

// ---------------------------------------------------------------------------
// Kernel 2: per-(b,c) greedy NMS. Column-chunked suppression bitmask in LDS.
// Area-sum term of the union computed per 16x16 tile with
// V_WMMA_F32_16X16X4_F32 (rank-2 outer sum: D[m][n] = area_m + area_n).
// fp contract OFF so inter=iw*ih is rounded before the union subtract,
// bit-matching the reference (no fma(-iw,ih,areasum) contraction).
// ---------------------------------------------------------------------------
__global__ void nms_suppress_kernel(const float* __restrict__ boxes,
                                    const int* __restrict__ order,
                                    const float* __restrict__ thr_p,
                                    unsigned* __restrict__ keep_out)
{
#pragma clang fp contract(off)
    __shared__ float sbx[NPAD], sby[NPAD], sbz[NPAD], sbw[NPAD]; // sorted boxes SoA
    __shared__ float sarea[NPAD];
    __shared__ unsigned supc[NPAD][8];   // sup bits for a 256-column chunk
    __shared__ unsigned keepw[32];       // 1000-bit keep mask

    const int bc  = blockIdx.x;
    const int b   = bc >> 5;             // C == 32
    const int tid = threadIdx.x;
    const float thresh = *thr_p;

    // Gather boxes in score order; pad with degenerate far-away zero-area boxes.
    for (int i = tid; i < NPAD; i += blockDim.x) {
        float x1, y1, x2, y2;
        if (i < NMS_N) {
            const int o = order[bc * ORD_STRIDE + i];
            const float* bp = boxes + ((size_t)b * NMS_N + o) * 4;
            x1 = bp[0]; y1 = bp[1]; x2 = bp[2]; y2 = bp[3];
        } else {
            x1 = y1 = x2 = y2 = 3.0e8f;  // inter=0 vs any real box, area=0
        }
        sbx[i] = x1; sby[i] = y1; sbz[i] = x2; sbw[i] = y2;
        sarea[i] = (x2 - x1) * (y2 - y1);
    }
    if (tid < 32) keepw[tid] = 0xFFFFFFFFu;
    __syncthreads();

    const int wv    = tid >> 5;          // 8 waves per block
    const int lane  = tid & 31;
    const int lhalf = lane >> 4;
    const int l15   = lane & 15;

    for (int chunk = 0; chunk < 4; ++chunk) {
        const int c0 = chunk * 256;

        for (int k = tid; k < NPAD * 8; k += blockDim.x)
            ((unsigned*)supc)[k] = 0u;
        __syncthreads();

        // 63 row-tiles x 16 col-tiles over this chunk; one wave per tile.
        for (int t = wv; t < 63 * 16; t += 8) {
            const int rt = t >> 4, ct = t & 15;
            const int r0 = rt * 16;
            const int cb = c0 + ct * 16;

            // WMMA operands (f32 16x16x4):
            //   A row m = [area_m, 1, 0, 0];  B col n = [1, area_n, 0, 0]
            // Lane-half selects K-half {0,1}|{2,3}; VGPR walks K within half.
            v2f a, bm;
            if (lhalf == 0) {
                a.x = sarea[r0 + l15]; a.y = 1.0f;             // A: K=0, K=1
                bm.x = 1.0f;           bm.y = sarea[cb + l15]; // B: K=0, K=1
            } else {
                a.x = 0.0f; a.y = 0.0f;                        // A: K=2, K=3
                bm.x = 0.0f; bm.y = 0.0f;                      // B: K=2, K=3
            }
            v8f acc = {};
            acc = __builtin_amdgcn_wmma_f32_16x16x4_f32(
                false, a, false, bm, (short)0, acc, false, false);
            // acc[v] for lane: element (m = r0+v+8*lhalf, n = cb+l15) = area_m+area_n

            const int n = cb + l15;
            const float cx1 = sbx[n], cy1 = sby[n], cx2 = sbz[n], cy2 = sbw[n];

#pragma unroll
            for (int v = 0; v < 8; ++v) {
                const int m = r0 + v + (lhalf << 3);
                const float lt_x = fmaxf(sbx[m], cx1);
                const float lt_y = fmaxf(sby[m], cy1);
                const float rb_x = fminf(sbz[m], cx2);
                const float rb_y = fminf(sbw[m], cy2);
                const float iw = fmaxf(rb_x - lt_x, 0.0f);
                const float ih = fmaxf(rb_y - lt_y, 0.0f);
                const float inter = iw * ih;                 // rounded once
                const float uni = acc[v] - inter;            // then subtracted
                const float iou = (uni > 0.0f) ? (inter / uni) : 0.0f;
                const bool pred = (iou > thresh) && (n > m) && (n < NMS_N);

                const unsigned msk = __builtin_amdgcn_ballot_w32(pred);
                if (lane == 0) {
                    const int w  = ct >> 1;           // word within chunk
                    const int sh = (ct & 1) * 16;     // 16-bit half of word
                    const unsigned lo = msk & 0xFFFFu;
                    const unsigned hi = msk >> 16;
                    if (lo) atomicOr(&supc[r0 + v][w],     lo << sh); // row r0+v
                    if (hi) atomicOr(&supc[r0 + v + 8][w], hi << sh); // row r0+v+8
                }
            }
        }
        __syncthreads();

        // Serial greedy scan: finalize keep bits for columns [c0, c0+256).
        if (tid == 0) {
            const int imax = (c0 + 256 < NMS_N) ? (c0 + 256) : NMS_N;
            const int w0 = c0 >> 5;
            for (int i = 0; i < imax; ++i) {
                if ((keepw[i >> 5] >> (i & 31)) & 1u) {
#pragma unroll
                    for (int w = 0; w < 8; ++w)
                        keepw[w0 + w] &= ~supc[i][w];
                }
            }
        }
        __syncthreads();
    }

    if (tid < 32) {
        unsigned v = keepw[tid];
        if (tid == 31) v &= 0xFFu;   // bits for i >= 1000 are invalid
        keep_out[bc * 32 + tid] = v;
    }
}

// ---------------------------------------------------------------------------
// Kernel 3: sequential cap/truncation + scatter of (b, c, box_idx) triples.
// L_b = min(L_{b-1} + kept_b, max_out*(b+1)); emitted count == final L <= B*max_out.
// ---------------------------------------------------------------------------
__global__ void nms_finalize_kernel(const int* __restrict__ order,
                                    const unsigned* __restrict__ keepg,
                                    const int* __restrict__ maxout_p,
                                    int* __restrict__ out, int out_elems)
{
    for (int k = threadIdx.x; k < out_elems; k += blockDim.x) out[k] = -1;
    __syncthreads();
    if (threadIdx.x != 0) return;

    const int maxout = *maxout_p;
    const int nrows  = out_elems / 3;
    int p = 0, L = 0;
    for (int b = 0; b < NMS_B; ++b) {
        const int cap = maxout * (b + 1);
        int ic = 0;                                  // valid count within batch
        for (int c = 0; c < NMS_C; ++c) {
            const int bc = b * NMS_C + c;
            for (int i = 0; i < NMS_N; ++i) {
                if ((keepg[bc * 32 + (i >> 5)] >> (i & 31)) & 1u) {
                    ++ic;
                    if (L + ic <= cap && p < nrows) {
                        out[p * 3 + 0] = b;
                        out[p * 3 + 1] = c;
                        out[p * 3 + 2] = order[bc * ORD_STRIDE + i];
                        ++p;
                    }
                }
            }
        }
        L = (L + ic < cap) ? (L + ic) : cap;
    }
}

// ---------------------------------------------------------------------------
extern "C" void kernel_launch(void* const* d_in, const int* in_sizes, int n_in,
                              void* d_out, int out_size, void* d_ws, size_t ws_size,
                              hipStream_t stream)
{
    const float* boxes  = (const float*)d_in[0];   // [B,N,4] f32
    const float* scores = (const float*)d_in[1];   // [B,C,N] f32
    const float* thr    = (const float*)d_in[2];   // scalar f32
    const int*   maxo   = (const int*)d_in[3];     // scalar i32
    int* out = (int*)d_out;                        // [B*max_out, 3] i32

    int*      order = (int*)d_ws;                                        // 1 MB
    unsigned* keepg = (unsigned*)((char*)d_ws +
                      (size_t)NMS_B * NMS_C * ORD_STRIDE * sizeof(int)); // 32 KB

    nms_sort_kernel    <<<NMS_B * NMS_C, 256, 0, stream>>>(scores, order);
    nms_suppress_kernel<<<NMS_B * NMS_C, 256, 0, stream>>>(boxes, order, thr, keepg);
    nms_finalize_kernel<<<1, 256, 0, stream>>>(order, keepg, maxo, out, out_size);
}